// MDCN_52424370815313
// MI455X (gfx1250) — compile-verified
//
#include <hip/hip_runtime.h>

#define B_N 2048
#define F_N 1024
#define M_N 128
#define KC  512                       // K-chunk staged in LDS (16 x 512 f32 = 32 KB)
#define RHO_1 0.95f
#define TAU_INV 1.0e-4f

typedef float v2f __attribute__((ext_vector_type(2)));
typedef float v8f __attribute__((ext_vector_type(8)));

// D = A(16x4 f32) * B(4x16 f32) + C(16x16 f32), wave32, full fp32 precision.
__device__ __forceinline__ v8f wmma_f32_16x16x4(v2f a, v2f b, v8f c) {
    return __builtin_amdgcn_wmma_f32_16x16x4_f32(false, a, false, b, (short)0, c,
                                                 false, false);
}

__global__ __launch_bounds__(256) void mdcn_fused_kernel(
    const float* __restrict__ feature,
    const float* __restrict__ rho_w, const float* __restrict__ rho_b,
    const float* __restrict__ pi_w,  const float* __restrict__ pi_b,
    const float* __restrict__ var_w, const float* __restrict__ var_b,
    const float* __restrict__ Wv,    const float* __restrict__ Zv,
    const float* __restrict__ muW,   const float* __restrict__ lvW,
    const float* __restrict__ lvZ,
    float* __restrict__ out_pi, float* __restrict__ out_mu,
    float* __restrict__ out_var)
{
    __shared__ float lds_A[16][KC];         // staged feature chunk (async TDM path)
    __shared__ float lds_out[3][16][M_N];   // raw GEMM results: rho/pi/var
    __shared__ float lds_dots[16][4];       // per-row dot products

    const int tid  = threadIdx.x;
    const int wave = tid >> 5;              // 0..7
    const int lane = tid & 31;
    const int hi   = lane >> 4;             // half-wave selector (K offset 0 or 2)
    const int l16  = lane & 15;
    const int rowTile = blockIdx.x;         // 16 batch rows per block

    // Low 32 bits of a generic LDS pointer == byte offset within wave's LDS.
    const unsigned ldsA_base = (unsigned)(uintptr_t)&lds_A[0][0];

    const float* slab  = feature + (size_t)rowTile * 16 * F_N;
    const float* bPtr0 = rho_w + (size_t)(wave * 16 + l16) * F_N;
    const float* bPtr1 = pi_w  + (size_t)(wave * 16 + l16) * F_N;
    const float* bPtr2 = var_w + (size_t)(wave * 16 + l16) * F_N;

    v8f acc0 = {0.f,0.f,0.f,0.f,0.f,0.f,0.f,0.f};
    v8f acc1 = {0.f,0.f,0.f,0.f,0.f,0.f,0.f,0.f};
    v8f acc2 = {0.f,0.f,0.f,0.f,0.f,0.f,0.f,0.f};

    const int drow  = tid >> 4;             // dot pass: 16 threads per row
    const int dpart = tid & 15;
    float s0 = 0.f, s1 = 0.f, s2 = 0.f;

    for (int c = 0; c < F_N / KC; ++c) {
        const int kbase = c * KC;

        // ---- async stage: 16 rows x KC f32 -> LDS (b128 per lane per issue)
        #pragma unroll
        for (int it = 0; it < (16 * KC) / (256 * 4); ++it) {   // 8 issues/thread
            const int j   = it * 256 + tid;                    // 16B chunk index
            const int r   = j >> 7;                            // KC/4 = 128 chunks/row
            const int k4  = (j & 127) << 2;                    // float idx in row
            const unsigned loff = ldsA_base + (unsigned)((r * KC + k4) * 4);
            const float* gp = slab + (size_t)r * F_N + kbase + k4;
            asm volatile("global_load_async_to_lds_b128 %0, %1, off"
                         :: "v"(loff), "v"((unsigned long long)(uintptr_t)gp)
                         : "memory");
        }
        asm volatile("s_wait_asynccnt 0x0" ::: "memory");
        __syncthreads();

        // ---- per-row dot partials from LDS (stride-16: conflict-free banks)
        #pragma unroll 4
        for (int i = 0; i < KC / 16; ++i) {
            const int kl = dpart + 16 * i;
            const int k  = kbase + kl;
            const float f  = lds_A[drow][kl];
            const float mw = muW[k];
            const float c1 = __expf(0.5f * (lvZ[k] - lvW[k])) * (Wv[k] - mw);
            s0 = fmaf(f, mw, s0);
            s1 = fmaf(f, c1, s1);
            s2 = fmaf(f, Zv[k], s2);
        }

        // ---- WMMA over chunk: A frag from LDS (b64), B frags from L2 (b64).
        // Half-wave K offset folded into addresses: no cndmask selects.
        #pragma unroll 4
        for (int kl = 0; kl < KC; kl += 4) {
            const int ko = kl + 2 * hi;
            float2 af = *(const float2*)&lds_A[l16][ko];
            v2f av; av.x = af.x; av.y = af.y;

            float2 bf = *(const float2*)(bPtr0 + kbase + ko);
            v2f bv; bv.x = bf.x; bv.y = bf.y;
            acc0 = wmma_f32_16x16x4(av, bv, acc0);

            bf = *(const float2*)(bPtr1 + kbase + ko);
            bv.x = bf.x; bv.y = bf.y;
            acc1 = wmma_f32_16x16x4(av, bv, acc1);

            bf = *(const float2*)(bPtr2 + kbase + ko);
            bv.x = bf.x; bv.y = bf.y;
            acc2 = wmma_f32_16x16x4(av, bv, acc2);
        }
        __syncthreads();   // all reads of lds_A done before next chunk overwrites
    }

    // ---- finish dot products: 16-lane shuffle tree, then park in LDS
    #pragma unroll
    for (int m = 8; m >= 1; m >>= 1) {
        s0 += __shfl_xor(s0, m, 16);
        s1 += __shfl_xor(s1, m, 16);
        s2 += __shfl_xor(s2, m, 16);
    }
    if (dpart == 0) {
        lds_dots[drow][0] = s0;
        lds_dots[drow][1] = s1;
        lds_dots[drow][2] = s2;
    }

    // ---- spill WMMA tiles. C/D layout: vgpr j -> M = j + 8*hi, N = l16.
    {
        const int ncol = wave * 16 + l16;
        #pragma unroll
        for (int j = 0; j < 8; ++j) {
            const int m = j + hi * 8;
            lds_out[0][m][ncol] = acc0[j];
            lds_out[1][m][ncol] = acc1[j];
            lds_out[2][m][ncol] = acc2[j];
        }
    }
    __syncthreads();

    // ---- elementwise epilogue + softmax: 16 threads/row, 8 strided cols each
    {
        const int row  = tid >> 4;
        const int part = tid & 15;
        const int grow = rowTile * 16 + row;
        const float dA = lds_dots[row][0];
        const float dB = lds_dots[row][1];
        const float dC = lds_dots[row][2];

        float plog[8];
        float pmax = -3.4e38f;
        #pragma unroll
        for (int t = 0; t < 8; ++t) {
            const int col = part + 16 * t;
            float rr  = lds_out[0][row][col] + rho_b[col];
            float rho = (col == 0) ? RHO_1 : tanhf(rr);

            float p = lds_out[1][row][col] + pi_b[col];
            plog[t] = p;
            pmax = fmaxf(pmax, p);

            float v0  = __expf(lds_out[2][row][col] + var_b[col]);
            float var = (1.f - __expf(rho)) * v0 + TAU_INV;

            float sq = 1.f - rho * rho;
            float mu = rho * dA + sqrtf(fmaxf(sq, 0.f)) * rho * dB + sq * dC;

            out_mu [(size_t)grow * M_N + col] = mu;
            out_var[(size_t)grow * M_N + col] = var;
        }
        #pragma unroll
        for (int m = 8; m >= 1; m >>= 1)
            pmax = fmaxf(pmax, __shfl_xor(pmax, m, 16));

        float pe[8];
        float psum = 0.f;
        #pragma unroll
        for (int t = 0; t < 8; ++t) {
            pe[t] = __expf(plog[t] - pmax);
            psum += pe[t];
        }
        #pragma unroll
        for (int m = 8; m >= 1; m >>= 1)
            psum += __shfl_xor(psum, m, 16);

        const float inv = 1.f / psum;
        #pragma unroll
        for (int t = 0; t < 8; ++t)
            out_pi[(size_t)grow * M_N + part + 16 * t] = pe[t] * inv;
    }
}

extern "C" void kernel_launch(void* const* d_in, const int* in_sizes, int n_in,
                              void* d_out, int out_size, void* d_ws, size_t ws_size,
                              hipStream_t stream) {
    (void)in_sizes; (void)n_in; (void)out_size; (void)d_ws; (void)ws_size;

    const float* feature = (const float*)d_in[0];
    const float* rho_w   = (const float*)d_in[1];
    const float* rho_b   = (const float*)d_in[2];
    const float* pi_w    = (const float*)d_in[3];
    const float* pi_b    = (const float*)d_in[4];
    const float* var_w   = (const float*)d_in[5];
    const float* var_b   = (const float*)d_in[6];
    const float* Wv      = (const float*)d_in[7];
    const float* Zv      = (const float*)d_in[8];
    const float* muW     = (const float*)d_in[9];
    const float* lvW     = (const float*)d_in[10];
    const float* lvZ     = (const float*)d_in[11];

    float* out_pi  = (float*)d_out;
    float* out_mu  = out_pi + (size_t)B_N * M_N;
    float* out_var = out_mu + (size_t)B_N * M_N;

    dim3 grid(B_N / 16), block(256);
    mdcn_fused_kernel<<<grid, block, 0, stream>>>(
        feature, rho_w, rho_b, pi_w, pi_b, var_w, var_b,
        Wv, Zv, muW, lvW, lvZ, out_pi, out_mu, out_var);
}